// CarafeConvolution_58205396796040
// MI455X (gfx1250) — compile-verified
//
#include <hip/hip_runtime.h>

typedef __attribute__((ext_vector_type(2))) float v2f;
typedef __attribute__((ext_vector_type(8))) float v8f;

// Problem constants (fixed by the reference harness)
#define NB   2      // batch
#define HH   128    // output H
#define WW   128    // output W
#define CC   128    // channels
#define hh   64     // lo-res h
#define ww   64     // lo-res w
#define KSZ  5      // kernel
#define K2   25     // kernel^2
#define COLS 12     // lo-res cols covered by a 16-pixel row tile: 8 + 2*pad
#define ROWS 5      // lo-res rows covered: kernel
#define KTOT 60     // ROWS*COLS, padded K of the per-tile GEMM (multiple of 4)
#define TPX  16     // output pixels per tile (WMMA M)

__global__ __launch_bounds__(128) void carafe_wmma_f32(
    const float* __restrict__ feat,   // [2,64,64,128]
    const float* __restrict__ masks,  // [2,128,128,25]
    float* __restrict__ out)          // [2,128,128,128]
{
  __shared__ float Flds[KTOT * CC];      // feature slab, [k = kr*12+kc][ch]
  __shared__ float Alds[TPX * KTOT];     // dense softmax-weight matrix [pixel][k]

  const int bid = blockIdx.x;
  const int xt  = bid & 7;               // tile index along W (W/16 = 8)
  const int i   = (bid >> 3) & 127;      // output row
  const int bb  = bid >> 10;             // batch
  const int t   = threadIdx.x;

  const int y   = i >> 1;                // lo-res source row (TF nearest, ratio 2)
  const int c0  = 8 * xt - 2;            // lo-res col base of the 12-col window

  // ---- Stage feature slab [5][12][128] -> LDS as [K=60][128], float4, zero OOB ----
  for (int f = t; f < KTOT * (CC / 4); f += 128) {
    const int pos = f >> 5;              // 32 float4 per (kr,kc) position
    const int q   = f & 31;
    const int kr  = pos / COLS;
    const int kc  = pos - kr * COLS;
    const int r   = y + kr - 2;
    const int c   = c0 + kc;
    float4 v = make_float4(0.f, 0.f, 0.f, 0.f);
    if (r >= 0 && r < hh && c >= 0 && c < ww) {
      const float4* gp =
          (const float4*)(feat + (((size_t)bb * hh + r) * ww + c) * CC) + q;
      v = *gp;
    }
    ((float4*)Flds)[pos * (CC / 4) + q] = v;
  }

  // ---- Zero A (holes of the scattered weight matrix must be 0) ----
  for (int f = t; f < TPX * KTOT; f += 128) Alds[f] = 0.f;
  __syncthreads();

  // ---- Softmax over 25 logits, scatter into dense A: A[p][di*12 + p/2 + dj] ----
  if (t < TPX) {
    const int j = 16 * xt + t;
    const float* mp = masks + (((size_t)bb * HH + i) * WW + j) * K2;
    float w[K2];
    float mx = -3.4e38f;
    #pragma unroll
    for (int q = 0; q < K2; ++q) { w[q] = mp[q]; mx = fmaxf(mx, w[q]); }
    float s = 0.f;
    #pragma unroll
    for (int q = 0; q < K2; ++q) { w[q] = __expf(w[q] - mx); s += w[q]; }
    const float inv = 1.f / s;
    const int base = t >> 1;             // p/2: window col offset of this pixel
    #pragma unroll
    for (int di = 0; di < KSZ; ++di)
      #pragma unroll
      for (int dj = 0; dj < KSZ; ++dj)
        Alds[t * KTOT + di * COLS + base + dj] = w[di * KSZ + dj] * inv;
  }
  __syncthreads();

  // ---- WMMA: Out[16 px][16 ch] = A[16x60] @ F[60x16ch], 15x v_wmma_f32_16x16x4_f32 ----
  const int lane = t & 31;               // wave32 lane
  const int wave = t >> 5;               // 4 waves per block
  const int hi   = lane >> 4;            // K/M split per ISA layout
  const int n    = lane & 15;            // B/D column (channel)
  const int m    = lane & 15;            // A row (pixel)

  for (int ct = wave; ct < CC / 16; ct += 4) {   // 2 channel tiles per wave
    const int nb = ct * 16;
    v8f acc = {};
    #pragma unroll
    for (int k0 = 0; k0 < KTOT; k0 += 4) {
      // A 16x4: VGPR0 = K (k0 + 2*hi), VGPR1 = K+1 -> contiguous, 8B-aligned pair
      v2f a = *(const v2f*)&Alds[m * KTOT + k0 + 2 * hi];
      // B 4x16: same K split, column n; rows stride CC in the slab
      v2f b;
      b.x = Flds[(k0 + 2 * hi)     * CC + nb + n];
      b.y = Flds[(k0 + 2 * hi + 1) * CC + nb + n];
      acc = __builtin_amdgcn_wmma_f32_16x16x4_f32(
          /*neg_a=*/false, a, /*neg_b=*/false, b,
          /*c_mod=*/(short)0, acc, /*reuse_a=*/false, /*reuse_b=*/false);
    }
    // D 16x16: VGPR r -> row (r + 8*hi), col n
    float* op = out + (((size_t)bb * HH + i) * WW + 16 * xt) * CC + nb + n;
    #pragma unroll
    for (int r = 0; r < 8; ++r) {
      const int M = r + 8 * hi;
      op[(size_t)M * CC] = acc[r];
    }
  }
}

extern "C" void kernel_launch(void* const* d_in, const int* in_sizes, int n_in,
                              void* d_out, int out_size, void* d_ws, size_t ws_size,
                              hipStream_t stream) {
  const float* feat  = (const float*)d_in[0];   // [2,64,64,128] fp32
  const float* masks = (const float*)d_in[1];   // [2,128,128,25] fp32
  float* out = (float*)d_out;                   // [2,128,128,128] fp32
  (void)in_sizes; (void)n_in; (void)out_size; (void)d_ws; (void)ws_size;

  const int blocks = NB * HH * (WW / TPX);      // 2048
  carafe_wmma_f32<<<blocks, 128, 0, stream>>>(feat, masks, out);
}